// GGCL_D_25537875542212
// MI455X (gfx1250) — compile-verified
//
#include <hip/hip_runtime.h>
#include <hip/hip_bf16.h>

typedef float v2f __attribute__((ext_vector_type(2)));
typedef float v8f __attribute__((ext_vector_type(8)));

#define IN_F 256
#define OUT_F 128

// ---------------------------------------------------------------------------
// Kernel A: zero the output accumulators (d_out is poisoned / stale between
// replays; the scatter kernel accumulates atomically).
// ---------------------------------------------------------------------------
__global__ void zero_kernel(float* __restrict__ p, long long n) {
    long long i = (long long)blockIdx.x * blockDim.x + threadIdx.x;
    long long stride = (long long)gridDim.x * blockDim.x;
    for (; i < n; i += stride) p[i] = 0.0f;
}

// ---------------------------------------------------------------------------
// Kernel B: fused dual GEMM + activations producing the per-node messages.
//   h_sig = relu(sigma @ Wsig); att = exp(-h_sig); m_sig = h_sig*att*att
//   h_mu  = elu(miu @ Wmu);                        m_mu  = h_mu*att
// One wave computes a 16-row x 128-col tile of BOTH outputs using
// V_WMMA_F32_16X16X4_F32 (fp32 tensor path, K stepped by 4).
// Block = 256 threads = 8 waves -> 128 rows per block.
// ---------------------------------------------------------------------------
__global__ __launch_bounds__(256)
void gemm_msg_kernel(const float* __restrict__ miu,
                     const float* __restrict__ sigma,
                     const float* __restrict__ Wmu,
                     const float* __restrict__ Wsig,
                     float* __restrict__ m_mu,
                     float* __restrict__ m_sig,
                     int N) {
    const int lane = threadIdx.x & 31;
    const int wave = threadIdx.x >> 5;
    const int row_base = blockIdx.x * 128 + wave * 16;

    const int m    = lane & 15;   // row-in-tile for A loads, col-in-tile for B/C
    const int kgrp = lane >> 4;   // 0: lanes 0-15, 1: lanes 16-31

    // Clamp so out-of-range waves still execute WMMA with valid addresses
    // (EXEC must remain all-ones for WMMA); stores are predicated instead.
    const int arow = (row_base + m < N) ? (row_base + m) : (N - 1);

    float att[8][8];

    // ================= sigma GEMM =================
    {
        v8f acc[8];
        v8f zero = {};
        #pragma unroll
        for (int t = 0; t < 8; ++t) acc[t] = zero;

        const float* Ap = sigma + (size_t)arow * IN_F + kgrp * 2;
        #pragma unroll 4
        for (int k0 = 0; k0 < IN_F; k0 += 4) {
            // A frag (16x4 f32): lane m holds A[m][k0+2*kgrp+{0,1}] -> contiguous float2
            v2f a = *(const v2f*)(Ap + k0);
            const float* Bp = Wsig + (size_t)(k0 + kgrp * 2) * OUT_F + m;
            #pragma unroll
            for (int t = 0; t < 8; ++t) {
                // B frag (4x16 f32): v0 = W[k][n], v1 = W[k+1][n], n = t*16+m
                v2f b;
                b[0] = Bp[t * 16];
                b[1] = Bp[t * 16 + OUT_F];
                acc[t] = __builtin_amdgcn_wmma_f32_16x16x4_f32(
                    false, a, false, b, (short)0, acc[t], false, false);
            }
        }
        // Epilogue: relu -> att -> m_sig
        #pragma unroll
        for (int t = 0; t < 8; ++t) {
            #pragma unroll
            for (int r = 0; r < 8; ++r) {
                float hs = fmaxf(acc[t][r], 0.0f);
                float a  = __expf(-hs);
                att[t][r] = a;
                int rr = row_base + r + kgrp * 8;   // C/D layout: M = r + 8*kgrp
                if (rr < N)
                    m_sig[(size_t)rr * OUT_F + t * 16 + m] = hs * a * a;
            }
        }
    }

    // ================= miu GEMM =================
    {
        v8f acc[8];
        v8f zero = {};
        #pragma unroll
        for (int t = 0; t < 8; ++t) acc[t] = zero;

        const float* Ap = miu + (size_t)arow * IN_F + kgrp * 2;
        #pragma unroll 4
        for (int k0 = 0; k0 < IN_F; k0 += 4) {
            v2f a = *(const v2f*)(Ap + k0);
            const float* Bp = Wmu + (size_t)(k0 + kgrp * 2) * OUT_F + m;
            #pragma unroll
            for (int t = 0; t < 8; ++t) {
                v2f b;
                b[0] = Bp[t * 16];
                b[1] = Bp[t * 16 + OUT_F];
                acc[t] = __builtin_amdgcn_wmma_f32_16x16x4_f32(
                    false, a, false, b, (short)0, acc[t], false, false);
            }
        }
        // Epilogue: elu -> * att -> m_mu
        #pragma unroll
        for (int t = 0; t < 8; ++t) {
            #pragma unroll
            for (int r = 0; r < 8; ++r) {
                float hm = acc[t][r];
                float e  = (hm > 0.0f) ? hm : (__expf(hm) - 1.0f);
                int rr = row_base + r + kgrp * 8;
                if (rr < N)
                    m_mu[(size_t)rr * OUT_F + t * 16 + m] = e * att[t][r];
            }
        }
    }
}

// ---------------------------------------------------------------------------
// Kernel C: edge scatter (SpMM). One wave32 per edge; lane handles 4 features
// via a b128 gather from the message arrays (L2-resident: 2x51MB < 192MB L2)
// and native f32 atomic adds into the outputs.
// ---------------------------------------------------------------------------
__global__ __launch_bounds__(256)
void spmm_scatter_kernel(const int* __restrict__ erow,
                         const int* __restrict__ ecol,
                         const float* __restrict__ a1,
                         const float* __restrict__ a2,
                         const float* __restrict__ m_mu,
                         const float* __restrict__ m_sig,
                         float* __restrict__ out_mu,
                         float* __restrict__ out_sig,
                         int E) {
    const int lane = threadIdx.x & 31;
    const long long wid = ((long long)blockIdx.x * blockDim.x + threadIdx.x) >> 5;
    if (wid >= E) return;
    const int e = (int)wid;

    const int r = erow[e];
    const int c = ecol[e];
    const float v1 = a1[e];
    const float v2 = a2[e];

    const float4 mm = ((const float4*)(m_mu  + (size_t)c * OUT_F))[lane];
    const float4 ms = ((const float4*)(m_sig + (size_t)c * OUT_F))[lane];

    float* om = out_mu  + (size_t)r * OUT_F + lane * 4;
    float* os = out_sig + (size_t)r * OUT_F + lane * 4;

    unsafeAtomicAdd(om + 0, v1 * mm.x);
    unsafeAtomicAdd(om + 1, v1 * mm.y);
    unsafeAtomicAdd(om + 2, v1 * mm.z);
    unsafeAtomicAdd(om + 3, v1 * mm.w);
    unsafeAtomicAdd(os + 0, v2 * ms.x);
    unsafeAtomicAdd(os + 1, v2 * ms.y);
    unsafeAtomicAdd(os + 2, v2 * ms.z);
    unsafeAtomicAdd(os + 3, v2 * ms.w);
}

// ---------------------------------------------------------------------------
extern "C" void kernel_launch(void* const* d_in, const int* in_sizes, int n_in,
                              void* d_out, int out_size, void* d_ws, size_t ws_size,
                              hipStream_t stream) {
    const float* miu   = (const float*)d_in[0];
    const float* sigma = (const float*)d_in[1];
    const int*   erow  = (const int*)  d_in[2];
    const int*   ecol  = (const int*)  d_in[3];
    const float* a1    = (const float*)d_in[4];
    const float* a2    = (const float*)d_in[5];
    const float* Wmu   = (const float*)d_in[6];
    const float* Wsig  = (const float*)d_in[7];

    const int N = in_sizes[0] / IN_F;
    const int E = in_sizes[2];

    float* out_mu  = (float*)d_out;
    float* out_sig = out_mu + (size_t)N * OUT_F;

    float* m_mu  = (float*)d_ws;
    float* m_sig = m_mu + (size_t)N * OUT_F;

    // 1) zero accumulators
    zero_kernel<<<2048, 256, 0, stream>>>(out_mu, (long long)2 * N * OUT_F);

    // 2) fused dual GEMM + activations -> messages (WMMA f32 path)
    gemm_msg_kernel<<<(N + 127) / 128, 256, 0, stream>>>(
        miu, sigma, Wmu, Wsig, m_mu, m_sig, N);

    // 3) edge scatter with native f32 atomics
    const long long waves = E;
    const long long blocks = (waves * 32 + 255) / 256;
    spmm_scatter_kernel<<<(int)blocks, 256, 0, stream>>>(
        erow, ecol, a1, a2, m_mu, m_sig, out_mu, out_sig, E);
}